// SwitchWhiten1d_12764642804233
// MI455X (gfx1250) — compile-verified
//
#include <hip/hip_runtime.h>
#include <hip/hip_bf16.h>

// ---------------------------------------------------------------------------
// SwitchWhiten1d on gfx1250 (MI455X), fp32 end-to-end.
//   N=16384 samples, C=2048 channels, groups of 16 channels -> 128 groups.
//   Phase 1: per-group Gram (16xN @ Nx16)  -> V_WMMA_F32_16X16X4_F32
//   Phase 2: Newton-Schulz on 16x16        -> V_WMMA_F32_16X16X4_F32 via LDS
//   Phase 3: y = A_g @ x + b_g (fused)     -> V_WMMA_F32_16X16X4_F32
// Memory-bound: ~402 MB of HBM traffic -> ~17 us at 23.3 TB/s; WMMA keeps
// the 2x537M-MAC matmuls off the VALU so we stay on the bandwidth roofline.
// ---------------------------------------------------------------------------

typedef __attribute__((ext_vector_type(2))) float v2f;
typedef __attribute__((ext_vector_type(8))) float v8f;

#define NROWS 16384
#define NCH   2048
#define CPG   16
#define NGRP  (NCH / CPG)                 // 128 groups
#define SLICES 16                         // K-slices for the Gram reduction
#define ROWS_PER_SLICE (NROWS / SLICES)   // 1024
#define CHUNKS (ROWS_PER_SLICE / 4)       // 256 WMMA k-steps (K=4 each)
#define NS_ITERS 5
#define EPSV 1e-5f

// D = A(16x4 f32) x B(4x16 f32) + C(16x16 f32)
__device__ __forceinline__ v8f wmma4(v2f a, v2f b, v8f c) {
  return __builtin_amdgcn_wmma_f32_16x16x4_f32(false, a, false, b, (short)0, c,
                                               false, false);
}

// ---------------------------------------------------------------------------
// Kernel 1: per-group partial Gram + per-channel partial sums.
// A-operand layout (fp32 16x4): lanes 0-15 M=0-15, lanes 16-31 M=0-15,
//   K = vgpr + 2*(lane>=16).  With A[m][k] = x[row_k][16g+m] and
//   B[k][n] = x[row_k][16g+n] the register images coincide -> D += X^T X.
// ---------------------------------------------------------------------------
__global__ __launch_bounds__(32) void cov_partial_kernel(
    const float* __restrict__ x, float* __restrict__ covp,
    float* __restrict__ sump) {
  const int lane = threadIdx.x;
  const int col = lane & 15;   // channel within group (both M and N index)
  const int hi  = lane >> 4;   // K-half selector
  const int g = blockIdx.x;
  const int s = blockIdx.y;

  const size_t colOff = (size_t)g * CPG + col;
  const float* p =
      x + ((size_t)s * ROWS_PER_SLICE + 2 * hi) * NCH + colOff;

  v8f acc = {0.f, 0.f, 0.f, 0.f, 0.f, 0.f, 0.f, 0.f};
  float csum = 0.0f;

  for (int c = 0; c < CHUNKS; ++c) {
    __builtin_prefetch(p + 16 * NCH, 0, 3);   // global_prefetch_b8
    float a0 = p[0];        // row (4c + 2*hi)     -> K = 0 + 2*hi
    float a1 = p[NCH];      // row (4c + 2*hi + 1) -> K = 1 + 2*hi
    v2f ab;
    ab.x = a0;
    ab.y = a1;
    acc = wmma4(ab, ab, acc);  // Gram accumulate: D[m][n] += sum_k x[k][m]x[k][n]
    csum += a0 + a1;
    p += 4 * NCH;
  }

  float* cp = covp + ((size_t)(g * SLICES + s) * 32 + lane) * 8;
#pragma unroll
  for (int i = 0; i < 8; ++i) cp[i] = acc[i];
  sump[(size_t)(g * SLICES + s) * 32 + lane] = csum;
}

// 16x16x16 matmul dst = X @ Y, all matrices row-major in LDS.
// Full-wave (EXEC all ones) as WMMA requires.
__device__ __forceinline__ void mat16_mul(float* dst, const float* X,
                                          const float* Y, int col, int hi) {
  v8f d = {0.f, 0.f, 0.f, 0.f, 0.f, 0.f, 0.f, 0.f};
#pragma unroll
  for (int q = 0; q < 4; ++q) {
    const int kb = 4 * q + 2 * hi;
    v2f a, b;
    a.x = X[col * 16 + kb];        // A[m=col][k=kb]
    a.y = X[col * 16 + kb + 1];    // A[m=col][k=kb+1]
    b.x = Y[kb * 16 + col];        // B[k=kb][n=col]
    b.y = Y[(kb + 1) * 16 + col];  // B[k=kb+1][n=col]
    d = wmma4(a, b, d);
  }
#pragma unroll
  for (int v = 0; v < 8; ++v) dst[(v + 8 * hi) * 16 + col] = d[v];
}

// ---------------------------------------------------------------------------
// Kernel 2: one wave per group. Reduce partials -> cov -> Newton-Schulz (WMMA)
// -> fused apply matrix W2[g] = weight * mw0 * sqrt(rTr) * P and
//    bconst[g][c] = bias - weight * mw0 * sqrt(rTr) * (P @ mean)[c].
// (softmax over 2 weights sums to 1, so (1 - mw1) == mw0.)
// ---------------------------------------------------------------------------
__global__ __launch_bounds__(32) void group_solve_kernel(
    const float* __restrict__ covp, const float* __restrict__ sump,
    const float* __restrict__ sw_mean_w, const float* __restrict__ sw_var_w,
    const float* __restrict__ weight, const float* __restrict__ bias,
    float* __restrict__ W2, float* __restrict__ bconst) {
  __shared__ float mP[256], mC[256], mT[256], mT2[256];
  __shared__ float red[32], meanS[16];

  const int lane = threadIdx.x;
  const int col = lane & 15;
  const int hi = lane >> 4;
  const int g = blockIdx.x;

  // ---- reduce per-channel sums -> mean ----
  float ssum = 0.f;
  for (int s = 0; s < SLICES; ++s)
    ssum += sump[(size_t)(g * SLICES + s) * 32 + lane];
  red[lane] = ssum;
  __syncthreads();
  const float mean_col = (red[col] + red[col + 16]) * (1.0f / NROWS);
  if (hi == 0) meanS[col] = mean_col;
  __syncthreads();

  // ---- reduce Gram partials (kept in WMMA D layout: M=v+8*hi, N=col) ----
  float cv[8];
#pragma unroll
  for (int v = 0; v < 8; ++v) {
    float t = 0.f;
    for (int s = 0; s < SLICES; ++s)
      t += covp[((size_t)(g * SLICES + s) * 32 + lane) * 8 + v];
    cv[v] = t;
  }

  const float vw0 = 1.0f / (1.0f + __expf(sw_var_w[1] - sw_var_w[0]));
  const float mw0 = 1.0f / (1.0f + __expf(sw_mean_w[1] - sw_mean_w[0]));

#pragma unroll
  for (int v = 0; v < 8; ++v) {
    const int m = v + 8 * hi;
    const float c_bn = cv[v] * (1.0f / NROWS) - meanS[m] * mean_col;
    mC[m * 16 + col] = vw0 * c_bn + ((m == col) ? EPSV : 0.0f);
  }
  __syncthreads();

  float tr = 0.f;
#pragma unroll
  for (int i = 0; i < 16; ++i) tr += mC[i * 17];
  const float rTr = 1.0f / tr;

#pragma unroll
  for (int v = 0; v < 8; ++v) {
    const int m = v + 8 * hi;
    mC[m * 16 + col] *= rTr;                     // covN
    mP[m * 16 + col] = (m == col) ? 1.0f : 0.0f; // P = I
  }
  __syncthreads();

  // ---- Newton-Schulz: P <- 1.5 P - 0.5 (P P P) covN ----
  for (int it = 0; it < NS_ITERS; ++it) {
    mat16_mul(mT, mP, mP, col, hi);   // P^2
    __syncthreads();
    mat16_mul(mT2, mT, mP, col, hi);  // P^3
    __syncthreads();
    mat16_mul(mT, mT2, mC, col, hi);  // P^3 @ covN
    __syncthreads();
#pragma unroll
    for (int v = 0; v < 8; ++v) {
      const int idx = (v + 8 * hi) * 16 + col;
      mP[idx] = 1.5f * mP[idx] - 0.5f * mT[idx];
    }
    __syncthreads();
  }

  const float srt = sqrtf(rTr);
  for (int e = lane; e < 256; e += 32) {
    const int c = e >> 4;
    W2[(size_t)g * 256 + e] = weight[g * CPG + c] * mw0 * srt * mP[e];
  }
  if (lane < 16) {
    float dot = 0.f;
#pragma unroll
    for (int d = 0; d < 16; ++d) dot += mP[lane * 16 + d] * meanS[d];
    bconst[g * CPG + lane] =
        bias[g * CPG + lane] - weight[g * CPG + lane] * mw0 * srt * dot;
  }
}

// ---------------------------------------------------------------------------
// Kernel 3: y = W2_g @ x + bconst_g. Each wave: B operands (whitening matrix)
// held in registers, 8 sample tiles of 16x16, x loaded as aligned float2
// pairs directly into A-operand layout, D stored coalesced with bias fused.
// ---------------------------------------------------------------------------
#define TILES_PER_WAVE 8
__global__ __launch_bounds__(256) void apply_kernel(
    const float* __restrict__ x, const float* __restrict__ W2,
    const float* __restrict__ bconst, float* __restrict__ y) {
  const int tid = threadIdx.x;
  const int lane = tid & 31;
  const int wave = tid >> 5;
  const int col = lane & 15;   // lanes 0-15 & 16-31 both cover M=0-15 / N=0-15
  const int hi = lane >> 4;
  const int g = blockIdx.x;
  const size_t gcol = (size_t)g * CPG;

  // B[k][n=c_out] = W2[c_out][k]; k = 4q + v + 2*hi
  const float* w2g = W2 + (size_t)g * 256;
  v2f b[4];
#pragma unroll
  for (int q = 0; q < 4; ++q) {
    const int kb = 4 * q + 2 * hi;
    b[q].x = w2g[col * 16 + kb];
    b[q].y = w2g[col * 16 + kb + 1];
  }
  const float bcv = bconst[gcol + col];

  const size_t rowBase =
      (size_t)blockIdx.y * 1024 + (size_t)wave * (TILES_PER_WAVE * 16);

  for (int t = 0; t < TILES_PER_WAVE; ++t) {
    const size_t row0 = rowBase + (size_t)t * 16;
    // A[m=col][k] = x[row0+col][16g + k]; aligned 8-byte pair per (q, vgpr)
    const float* xr = x + (row0 + col) * NCH + gcol;
    __builtin_prefetch(xr + 16 * NCH, 0, 3);
    v2f a[4];
#pragma unroll
    for (int q = 0; q < 4; ++q)
      a[q] = *(const v2f*)(xr + 4 * q + 2 * hi);

    v8f d = {0.f, 0.f, 0.f, 0.f, 0.f, 0.f, 0.f, 0.f};
#pragma unroll
    for (int q = 0; q < 4; ++q) d = wmma4(a[q], b[q], d);

    // D element (M = v + 8*hi, N = col) -> y[row0 + M][16g + col]
    float* yr = y + (row0 + 8 * hi) * NCH + gcol + col;
#pragma unroll
    for (int v = 0; v < 8; ++v) yr[(size_t)v * NCH] = d[v] + bcv;
  }
}

// ---------------------------------------------------------------------------
extern "C" void kernel_launch(void* const* d_in, const int* in_sizes, int n_in,
                              void* d_out, int out_size, void* d_ws,
                              size_t ws_size, hipStream_t stream) {
  const float* x = (const float*)d_in[0];
  const float* sw_mean_w = (const float*)d_in[1];
  const float* sw_var_w = (const float*)d_in[2];
  const float* weight = (const float*)d_in[3];
  const float* bias = (const float*)d_in[4];
  float* y = (float*)d_out;

  // workspace layout (floats): deterministic two-stage reduction, no atomics
  float* ws = (float*)d_ws;
  float* covp = ws;                                    // 128*16*32*8 = 524288
  float* sump = covp + (size_t)NGRP * SLICES * 32 * 8; // 128*16*32   =  65536
  float* W2 = sump + (size_t)NGRP * SLICES * 32;       // 128*256     =  32768
  float* bconst = W2 + (size_t)NGRP * 256;             // 128*16      =   2048

  cov_partial_kernel<<<dim3(NGRP, SLICES), 32, 0, stream>>>(x, covp, sump);
  group_solve_kernel<<<dim3(NGRP), 32, 0, stream>>>(
      covp, sump, sw_mean_w, sw_var_w, weight, bias, W2, bconst);
  apply_kernel<<<dim3(NGRP, NROWS / 1024), 256, 0, stream>>>(x, W2, bconst, y);
}